// MultiheadSelfAttention_45432164057644
// MI455X (gfx1250) — compile-verified
//
#include <hip/hip_runtime.h>
#include <hip/hip_bf16.h>

typedef __attribute__((ext_vector_type(2))) float v2f;
typedef __attribute__((ext_vector_type(8))) float v8f;

#define B_      32
#define E_      4096
#define SPLIT_E_ 1024
#define HPS_    8
#define HD_     128
#define NH_     256      // B_*HPS_
#define S_      2048
#define NEG_    (-10000.0f)
#define CH_     8        // s-chunks per head
#define CS_     256      // S_/CH_
#define KSPLIT_ 4
#define KCH_    1024     // E_/KSPLIT_

static __device__ __forceinline__ v8f wmma_f32(v2f a, v2f b, v8f c) {
  // D = A(16x4) * B(4x16) + C, fp32 WMMA (wave32)
  return __builtin_amdgcn_wmma_f32_16x16x4_f32(false, a, false, b, (short)0, c,
                                               false, false);
}

// ---------------- Kernel 1: QKV projection (WMMA f32, K split 4-way) --------
// part layout: [mat(3)][ks(4)][row 32][col 1024]
__global__ void qkv_proj_wmma(const float* __restrict__ x,
                              const float* __restrict__ wq,
                              const float* __restrict__ wk,
                              const float* __restrict__ wv,
                              float* __restrict__ part) {
  const int tid  = threadIdx.x;
  const int lane = tid & 31;
  const int wg   = blockIdx.x * 4 + (tid >> 5);   // 0..767 waves
  const int mat  = wg >> 8;                       // 0..2
  const int rem  = wg & 255;
  const int strip = rem >> 2;                     // 0..63 -> 16-col strip
  const int ks    = rem & 3;                      // K chunk
  const int j0    = strip * 16;
  const int kbase = ks * KCH_;
  const float* __restrict__ W = (mat == 0) ? wq : (mat == 1 ? wk : wv);
  const int m    = lane & 15;                     // A row / B col within tile
  const int koff = (lane < 16) ? 0 : 2;           // K pair select
  v8f acc0 = {};  // rows 0..15
  v8f acc1 = {};  // rows 16..31
#pragma unroll 4
  for (int k0 = kbase; k0 < kbase + KCH_; k0 += 4) {
    v2f a0 = *(const v2f*)(x + m * E_ + k0 + koff);
    v2f a1 = *(const v2f*)(x + (m + 16) * E_ + k0 + koff);
    v2f bf;
    bf.x = W[(size_t)(k0 + koff)     * SPLIT_E_ + j0 + m];
    bf.y = W[(size_t)(k0 + koff + 1) * SPLIT_E_ + j0 + m];
    acc0 = wmma_f32(a0, bf, acc0);
    acc1 = wmma_f32(a1, bf, acc1);
  }
  float* p = part + (size_t)(mat * KSPLIT_ + ks) * B_ * SPLIT_E_;
  const int col   = j0 + m;
  const int rbase = (lane < 16) ? 0 : 8;
#pragma unroll
  for (int i = 0; i < 8; ++i) {
    p[(rbase + i)      * SPLIT_E_ + col] = acc0[i];
    p[(rbase + i + 16) * SPLIT_E_ + col] = acc1[i];
  }
}

// ---------------- Kernel 2: reduce K-split partials + bias ------------------
__global__ void qkv_reduce(const float* __restrict__ part,
                           const float* __restrict__ bq,
                           const float* __restrict__ bk,
                           const float* __restrict__ bv,
                           float* __restrict__ q_ws,
                           float* __restrict__ k_new,
                           float* __restrict__ v_new) {
  const int idx = blockIdx.x * blockDim.x + threadIdx.x;
  if (idx >= 3 * B_ * SPLIT_E_) return;
  const int mat = idx / (B_ * SPLIT_E_);
  const int r   = idx - mat * (B_ * SPLIT_E_);
  const int col = r & (SPLIT_E_ - 1);
  float s = 0.f;
#pragma unroll
  for (int ks = 0; ks < KSPLIT_; ++ks)
    s += part[(size_t)(mat * KSPLIT_ + ks) * B_ * SPLIT_E_ + r];
  const float* bias = (mat == 0) ? bq : (mat == 1 ? bk : bv);
  s += bias[col];
  float* dst = (mat == 0) ? q_ws : (mat == 1 ? k_new : v_new);
  dst[r] = s;  // flat [b*8+h][128] == flat [b][1024]
}

// ---------------- Kernel 3: flash-decode partials per (head, s-chunk) ------
// v rows are async-staged into LDS (ASYNCcnt) overlapping the k score stream.
// part layout per (n,c): [m, lsum, acc[128]]
__global__ void attn_partial(const float* __restrict__ q_ws,
                             const float* __restrict__ k_cache,
                             const float* __restrict__ v_cache,
                             const float* __restrict__ k_new,
                             const float* __restrict__ v_new,
                             const unsigned char* __restrict__ mask,
                             float* __restrict__ part) {
  __shared__ float v_sh[CS_ * HD_];   // 128 KB staging (320KB/WGP -> 2 WGs)
  __shared__ float q_sh[HD_];
  __shared__ float p_sh[CS_];
  __shared__ float red_sh[CS_];
  __shared__ float acc_sh[HD_];
  const int tid  = threadIdx.x;           // 256 threads = 8 waves
  const int lane = tid & 31;
  const int wvid = tid >> 5;
  const int n = blockIdx.x / CH_;         // head = b*8+h
  const int c = blockIdx.x % CH_;
  const int b = n / HPS_;
  if (tid < HD_) { q_sh[tid] = q_ws[n * HD_ + tid]; acc_sh[tid] = 0.f; }

  // ---- async-stage this wave's 32 v rows into LDS (overlaps score pass) ---
  // One coalesced 512B global_load_async_to_lds_b128 per row per wave.
  {
#pragma unroll 8
    for (int i = 0; i < 32; ++i) {
      const int sl  = wvid * 32 + i;
      const int sg2 = c * CS_ + sl;
      const float* vrow = (sg2 == S_ - 1)
                              ? (v_new + n * HD_)
                              : (v_cache + ((size_t)sg2 * NH_ + n) * HD_);
      const unsigned long long ga =
          (unsigned long long)(uintptr_t)vrow + (unsigned)(lane * 16);
      // low 32 bits of the flat shared-aperture address == LDS byte offset
      const unsigned lo =
          (unsigned)(uintptr_t)(v_sh + sl * HD_) + (unsigned)(lane * 16);
      asm volatile("global_load_async_to_lds_b128 %0, %1, off"
                   :: "v"(lo), "v"(ga)
                   : "memory");
    }
  }
  __syncthreads();  // q_sh ready

  // ---- scores: one s per thread, per-lane streaming of a 512B k row ------
  const int sg = c * CS_ + tid;
  const float* krow = (sg == S_ - 1) ? (k_new + n * HD_)
                                     : (k_cache + ((size_t)sg * NH_ + n) * HD_);
  float dot = 0.f;
#pragma unroll
  for (int d0 = 0; d0 < HD_; d0 += 4) {
    const float4 kk = *(const float4*)(krow + d0);
    dot += kk.x * q_sh[d0] + kk.y * q_sh[d0 + 1] +
           kk.z * q_sh[d0 + 2] + kk.w * q_sh[d0 + 3];
  }
  const float score = mask[b * S_ + sg] ? dot : NEG_;
  p_sh[tid]   = score;
  red_sh[tid] = score;
  __syncthreads();
  for (int st = CS_ / 2; st > 0; st >>= 1) {
    if (tid < st) red_sh[tid] = fmaxf(red_sh[tid], red_sh[tid + st]);
    __syncthreads();
  }
  const float m = red_sh[0];
  __syncthreads();
  const float pr = __expf(score - m);
  p_sh[tid]   = pr;
  red_sh[tid] = pr;
  __syncthreads();
  for (int st = CS_ / 2; st > 0; st >>= 1) {
    if (tid < st) red_sh[tid] += red_sh[tid + st];
    __syncthreads();
  }
  const float lsum = red_sh[0];

  // ---- weighted V from LDS: wave reads only the rows it staged -----------
  asm volatile("s_wait_asynccnt 0x0" ::: "memory");
  float ax = 0.f, ay = 0.f, az = 0.f, aw = 0.f;
  const int d = lane * 4;
  for (int i = 0; i < 32; ++i) {
    const int sl = wvid * 32 + i;
    const float4 vv = *(const float4*)(v_sh + sl * HD_ + d);  // ds_load_b128
    const float p = p_sh[sl];
    ax += p * vv.x; ay += p * vv.y; az += p * vv.z; aw += p * vv.w;
  }
  atomicAdd(&acc_sh[d],     ax);   // ds_add_f32
  atomicAdd(&acc_sh[d + 1], ay);
  atomicAdd(&acc_sh[d + 2], az);
  atomicAdd(&acc_sh[d + 3], aw);
  __syncthreads();
  float* pp = part + (size_t)(n * CH_ + c) * (HD_ + 2);
  if (tid == 0) { pp[0] = m; pp[1] = lsum; }
  if (tid < HD_) pp[2 + tid] = acc_sh[tid];
}

// ---------------- Kernel 4: combine chunk partials --------------------------
__global__ void attn_combine(const float* __restrict__ part,
                             float* __restrict__ val_ws) {
  const int n   = blockIdx.x;   // 256 heads
  const int tid = threadIdx.x;  // 128 = HD_
  __shared__ float mm[CH_], ll[CH_];
  if (tid < CH_) {
    mm[tid] = part[(size_t)(n * CH_ + tid) * (HD_ + 2)];
    ll[tid] = part[(size_t)(n * CH_ + tid) * (HD_ + 2) + 1];
  }
  __syncthreads();
  float M = -3.4e38f;
#pragma unroll
  for (int cc = 0; cc < CH_; ++cc) M = fmaxf(M, mm[cc]);
  float L = 0.f, v = 0.f;
#pragma unroll
  for (int cc = 0; cc < CH_; ++cc) {
    const float w = __expf(mm[cc] - M);
    L += ll[cc] * w;
    v += w * part[(size_t)(n * CH_ + cc) * (HD_ + 2) + 2 + tid];
  }
  val_ws[n * HD_ + tid] = v / L;  // flat [b*8+h][128] == flat [b][1024]
}

// ---------------- Kernel 5: output projection (WMMA f32) + residual --------
__global__ void out_proj_wmma(const float* __restrict__ val_ws,
                              const float* __restrict__ w_out,
                              const float* __restrict__ b_out,
                              const float* __restrict__ x,
                              float* __restrict__ out) {
  const int tid  = threadIdx.x;
  const int lane = tid & 31;
  const int wg   = blockIdx.x * 4 + (tid >> 5);  // 0..255 -> 16-col strip of E
  const int e0   = wg * 16;
  const int m    = lane & 15;
  const int koff = (lane < 16) ? 0 : 2;
  v8f acc0 = {};
  v8f acc1 = {};
#pragma unroll 4
  for (int j0 = 0; j0 < SPLIT_E_; j0 += 4) {
    v2f a0 = *(const v2f*)(val_ws + m * SPLIT_E_ + j0 + koff);
    v2f a1 = *(const v2f*)(val_ws + (m + 16) * SPLIT_E_ + j0 + koff);
    // B[k][n] = w_out[(e0+n)*1024 + j0+k] -> contiguous b64 along K
    v2f bf = *(const v2f*)(w_out + (size_t)(e0 + m) * SPLIT_E_ + j0 + koff);
    acc0 = wmma_f32(a0, bf, acc0);
    acc1 = wmma_f32(a1, bf, acc1);
  }
  const int col   = e0 + m;
  const int rbase = (lane < 16) ? 0 : 8;
#pragma unroll
  for (int i = 0; i < 8; ++i) {
    const int r0 = rbase + i;
    out[r0 * E_ + col] = acc0[i] + b_out[col] + x[r0 * E_ + col];
    const int r1 = r0 + 16;
    out[r1 * E_ + col] = acc1[i] + b_out[col] + x[r1 * E_ + col];
  }
}

extern "C" void kernel_launch(void* const* d_in, const int* in_sizes, int n_in,
                              void* d_out, int out_size, void* d_ws, size_t ws_size,
                              hipStream_t stream) {
  const float* x  = (const float*)d_in[0];
  const float* wq = (const float*)d_in[1];
  const float* bq = (const float*)d_in[2];
  const float* wk = (const float*)d_in[3];
  const float* bk = (const float*)d_in[4];
  const float* wv = (const float*)d_in[5];
  const float* bv = (const float*)d_in[6];
  const float* kc = (const float*)d_in[7];
  const float* vc = (const float*)d_in[8];
  const float* wo = (const float*)d_in[9];
  const float* bo = (const float*)d_in[10];
  const unsigned char* mask = (const unsigned char*)d_in[11];  // jax bool = 1B
  (void)in_sizes; (void)n_in; (void)out_size; (void)ws_size;

  float* out  = (float*)d_out;           // [32, 4096]
  float* knew = out + B_ * E_;           // [256, 128]
  float* vnew = knew + NH_ * HD_;        // [256, 128]

  float* ws     = (float*)d_ws;
  float* q_ws   = ws;                                   // 32768 floats
  float* val_ws = ws + 32768;                           // 32768 floats
  float* attn_p = ws + 65536;                           // 256*8*130 = 266240
  float* qkv_p  = ws + 65536 + NH_ * CH_ * (HD_ + 2);   // 3*4*32768 = 393216

  qkv_proj_wmma<<<192, 128, 0, stream>>>(x, wq, wk, wv, qkv_p);
  qkv_reduce<<<(3 * B_ * SPLIT_E_ + 255) / 256, 256, 0, stream>>>(
      qkv_p, bq, bk, bv, q_ws, knew, vnew);
  attn_partial<<<NH_ * CH_, CS_, 0, stream>>>(q_ws, kc, vc, knew, vnew, mask,
                                              attn_p);
  attn_combine<<<NH_, HD_, 0, stream>>>(attn_p, val_ws);
  out_proj_wmma<<<64, 128, 0, stream>>>(val_ws, wo, bo, x, out);
}